// FixedGCNModel_83562883711396
// MI455X (gfx1250) — compile-verified
//
#include <hip/hip_runtime.h>

typedef float v2f __attribute__((ext_vector_type(2)));
typedef float v8f __attribute__((ext_vector_type(8)));

#define NGRAPH 8
#define NNODES 50000
#define FDIM   128

// -------------------- utility kernels --------------------

__global__ __launch_bounds__(256) void k_zero(float* __restrict__ p, int n) {
  int i = blockIdx.x * blockDim.x + threadIdx.x;
  if (i < n) p[i] = 0.0f;
}

__global__ __launch_bounds__(256) void k_deg(const int* __restrict__ dst,
                                             float* __restrict__ deg, int E) {
  int e = blockIdx.x * blockDim.x + threadIdx.x;
  if (e < E) atomicAdd(deg + dst[e], 1.0f);
}

__global__ __launch_bounds__(256) void k_dinv(float* __restrict__ deg, int n) {
  int i = blockIdx.x * blockDim.x + threadIdx.x;
  if (i < n) deg[i] = rsqrtf(deg[i] + 1.0f);   // +1 self-loop
}

// -------------------- GEMM: Hw = Hcur @ W  (exact fp32 WMMA) --------------------
// grid.x = NNODES/16 row tiles; 256 threads = 8 waves; wave w -> columns [16w,16w+16)
__global__ __launch_bounds__(256) void k_gemm(const float* __restrict__ A,
                                              const float* __restrict__ B,
                                              float* __restrict__ C) {
  const int lane = threadIdx.x & 31;
  const int wave = threadIdx.x >> 5;
  const int row0 = blockIdx.x << 4;
  const int col0 = wave << 4;
  const int n    = lane & 15;                 // M index for A frag, N index for B frag
  const int kb   = (lane >> 4) << 1;          // 0 or 2 (K sub-offset per ISA layout)

  v8f acc = {};
  const float* ap = A + (size_t)(row0 + n) * FDIM + kb;
  const float* bp = B + (size_t)kb * FDIM + col0 + n;

  for (int k4 = 0; k4 < FDIM; k4 += 4) {
    v2f a, b;
    a.x = ap[k4 + 0];
    a.y = ap[k4 + 1];
    b.x = bp[(size_t)k4 * FDIM];              // W[k4+kb][col0+n]
    b.y = bp[(size_t)(k4 + 1) * FDIM];        // W[k4+kb+1][col0+n]
    // D = A*B + C ; emits v_wmma_f32_16x16x4_f32
    acc = __builtin_amdgcn_wmma_f32_16x16x4_f32(false, a, false, b,
                                                (short)0, acc, false, false);
  }

  // D layout: VGPR v -> row (row0 + v [+8 for lanes 16..31]), col = col0 + (lane&15)
  const int rbase = row0 + ((lane >> 4) << 3);
  float* cp = C + (size_t)rbase * FDIM + col0 + n;
#pragma unroll
  for (int v = 0; v < 8; ++v) cp[(size_t)v * FDIM] = acc[v];
}

// -------------------- edge scatter: agg[dst] += Hw[src] * dinv[src] --------------------
// one wave per edge: 32 lanes x float4 = 128 features
__global__ __launch_bounds__(256) void k_scatter(const float* __restrict__ Hw,
                                                 const float* __restrict__ dinv,
                                                 const int* __restrict__ src,
                                                 const int* __restrict__ dst,
                                                 float* __restrict__ agg, int E) {
  long long t = (long long)blockIdx.x * blockDim.x + threadIdx.x;
  if (t >= (long long)E * 32) return;
  const int e = (int)(t >> 5);
  const int q = (int)(t & 31);
  const int s = src[e];
  const int d = dst[e];
  const float sc = dinv[s];
  const float4 v = *(const float4*)(Hw + (size_t)s * FDIM + (q << 2));
  float* out = agg + (size_t)d * FDIM + (q << 2);
  atomicAdd(out + 0, v.x * sc);
  atomicAdd(out + 1, v.y * sc);
  atomicAdd(out + 2, v.z * sc);
  atomicAdd(out + 3, v.w * sc);
}

// -------------------- combine: out = relu(agg*dinv + Hw*dinv^2 + b) --------------------
__global__ __launch_bounds__(256) void k_combine(const float4* __restrict__ agg,
                                                 const float4* __restrict__ Hw,
                                                 const float* __restrict__ dinv,
                                                 const float4* __restrict__ bias,
                                                 float4* __restrict__ out, int nq) {
  int i = blockIdx.x * blockDim.x + threadIdx.x;
  if (i >= nq) return;
  const int node = i >> 5;                    // 32 float4 per node
  const float di = dinv[node];
  const float d2 = di * di;
  const float4 a = agg[i];
  const float4 h = Hw[i];
  const float4 b = bias[i & 31];
  float4 r;
  r.x = fmaxf(fmaf(a.x, di, fmaf(h.x, d2, b.x)), 0.0f);
  r.y = fmaxf(fmaf(a.y, di, fmaf(h.y, d2, b.y)), 0.0f);
  r.z = fmaxf(fmaf(a.z, di, fmaf(h.z, d2, b.z)), 0.0f);
  r.w = fmaxf(fmaf(a.w, di, fmaf(h.w, d2, b.w)), 0.0f);
  out[i] = r;
}

// -------------------- host-side orchestration --------------------

extern "C" void kernel_launch(void* const* d_in, const int* in_sizes, int n_in,
                              void* d_out, int out_size, void* d_ws, size_t ws_size,
                              hipStream_t stream) {
  const float* x  = (const float*)d_in[0];
  const int*   ei = (const int*)d_in[1];                 // [G, 2, E] (int32 under jax defaults)
  const float* W[3] = {(const float*)d_in[2], (const float*)d_in[4], (const float*)d_in[6]};
  const float* B[3] = {(const float*)d_in[3], (const float*)d_in[5], (const float*)d_in[7]};

  const int N = NNODES, F = FDIM;
  const int E = in_sizes[1] / (2 * NGRAPH);              // 800000
  const size_t NF = (size_t)N * F;

  // workspace layout (per-graph working set fits in L2: keeps scatter traffic off HBM)
  char* ws = (char*)d_ws;
  float* dinv = (float*)ws;                              // N floats
  size_t off = ((size_t)N * 4 + 255) & ~(size_t)255;
  float* Hw   = (float*)(ws + off);                      // N*F
  float* agg  = (float*)(ws + off + NF * 4);             // N*F
  float* bufA = (float*)(ws + off + 2 * NF * 4);         // N*F

  const int BLK = 256;
  const int gN   = (N + BLK - 1) / BLK;
  const int gE   = (E + BLK - 1) / BLK;
  const int gNF  = (int)((NF + BLK - 1) / BLK);
  const int gSc  = (int)(((long long)E * 32 + BLK - 1) / BLK);
  const int gCmb = (int)((NF / 4 + BLK - 1) / BLK);
  const int gGemm = N / 16;                              // 3125, exact

  for (int g = 0; g < NGRAPH; ++g) {
    const int* src = ei + (size_t)g * 2 * E;
    const int* dst = src + E;

    // degree -> dinv (in place), shared by all 3 layers
    k_zero<<<gN, BLK, 0, stream>>>(dinv, N);
    k_deg<<<gE, BLK, 0, stream>>>(dst, dinv, E);
    k_dinv<<<gN, BLK, 0, stream>>>(dinv, N);

    const float* Hcur = x + (size_t)g * NF;
    for (int l = 0; l < 3; ++l) {
      k_gemm<<<gGemm, BLK, 0, stream>>>(Hcur, W[l], Hw);
      k_zero<<<gNF, BLK, 0, stream>>>(agg, (int)NF);
      k_scatter<<<gSc, BLK, 0, stream>>>(Hw, dinv, src, dst, agg, E);
      float* outp = (l == 2) ? ((float*)d_out + (size_t)g * NF) : bufA;
      k_combine<<<gCmb, BLK, 0, stream>>>((const float4*)agg, (const float4*)Hw, dinv,
                                          (const float4*)B[l], (float4*)outp, (int)(NF / 4));
      Hcur = bufA;
    }
  }
}